// Net2_33835752358576
// MI455X (gfx1250) — compile-verified
//
#include <hip/hip_runtime.h>
#include <math.h>

typedef __attribute__((ext_vector_type(2))) float v2f;
typedef __attribute__((ext_vector_type(8))) float v8f;

#define NUM_PTS 1000

// ---------------------------------------------------------------------------
// Fused fp32-WMMA MLP layer: out[M x 128] = relu(in[M x K] @ W^T + bias)
//   in : M x K   row-major
//   W  : 128 x K row-major (PyTorch/JAX Linear convention: W[out][in])
// Each wave computes one 16x16 output tile with V_WMMA_F32_16X16X4_F32,
// stepping K by 4. 8 waves per block cover all 128 output columns.
// A-fragment layout (ISA 7.12.2, 32-bit A 16x4): lane = (k>=2?16:0)+m,
//   vgpr0 holds K = 0/2, vgpr1 holds K = 1/3.
// B-fragment (4x16 KxN): lane = (k>=2?16:0)+n, same vgpr split over K.
// C/D 16x16 f32: vgpr r -> row r (lanes 0-15) / row r+8 (lanes 16-31).
// Store path: single scalar full-tile test -> straight-line stores in the
// common (tile-aligned) case; guarded path only for a ragged final tile.
// ---------------------------------------------------------------------------
template <int K>
__global__ __launch_bounds__(256) void mlp_layer_wmma(
    const float* __restrict__ in, const float* __restrict__ W,
    const float* __restrict__ bias, float* __restrict__ out, int M) {
  const int lane = threadIdx.x & 31;
  const int wave = threadIdx.x >> 5;
  const int row0 = blockIdx.x << 4;
  const int col0 = wave << 4;
  const int l15  = lane & 15;
  const int hi   = lane >> 4;            // 0: lanes 0-15, 1: lanes 16-31
  int mrow = row0 + l15;
  if (mrow >= M) mrow = M - 1;           // clamp (dup row, stores guarded)
  const int ncol  = col0 + l15;
  const int khalf = hi * 2;              // K offset 0 or 2 within a step

  v8f c = {};
  for (int k0 = 0; k0 < K; k0 += 4) {
    const int ka = k0 + khalf;
    v2f a, b;
    a.x = in[mrow * K + ka + 0];
    a.y = in[mrow * K + ka + 1];
    b.x = W[ncol * K + ka + 0];
    b.y = W[ncol * K + ka + 1];
    // args: (neg_a, A, neg_b, B, c_mod, C, reuse_a, reuse_b)
    c = __builtin_amdgcn_wmma_f32_16x16x4_f32(false, a, false, b, (short)0, c,
                                              false, false);
  }

  const float bv    = bias[ncol];
  const int   mbase = row0 + hi * 8;
  if (row0 + 16 <= M) {                  // uniform: full tile, no guards
#pragma unroll
    for (int r = 0; r < 8; ++r) {
      float v = c[r] + bv;
      out[(mbase + r) * 128 + ncol] = v > 0.0f ? v : 0.0f;
    }
  } else {                               // cold ragged-tile fallback
#pragma unroll
    for (int r = 0; r < 8; ++r) {
      const int m = mbase + r;
      if (m < M) {
        float v = c[r] + bv;
        out[m * 128 + ncol] = v > 0.0f ? v : 0.0f;
      }
    }
  }
}

// ---------------------------------------------------------------------------
// Cox–de-Boor basis row at parameter t for the 12-knot vector (degree 3).
// Mirrors reference construct(): N has 11 entries, ascending in-place update
// reads old N[i+1] before it is overwritten -> matches functional update.
// ---------------------------------------------------------------------------
__device__ __forceinline__ void construct_row(float t, const float* kn,
                                              float* N) {
#pragma unroll
  for (int i = 0; i < 11; ++i)
    N[i] = (t >= kn[i] && t < kn[i + 1]) ? 1.0f : 0.0f;
#pragma unroll
  for (int p = 1; p <= 3; ++p) {
    const int m = 10 - p;
    for (int i = 0; i < m; ++i) {
      const float da = kn[p + i] - kn[i];
      const float a  = (da != 0.0f) ? (t - kn[i]) / da : 0.0f;
      const float db = kn[p + 1 + i] - kn[1 + i];
      const float bb = (db != 0.0f) ? (kn[p + 1 + i] - t) / db : 0.0f;
      N[i] = N[i] * a + N[i + 1] * bb;
    }
  }
}

// ---------------------------------------------------------------------------
// Per-batch: layer3 + softmax + cumsum -> knots; build A(8x8); solve A C = P
// (Gauss-Jordan, partial pivot, 2 RHS); derivative control points Q(7), R(6).
// Writes knots into d_out and [knots|Qx|Qy|Rx|Ry] (stride 40) into ws.
// ---------------------------------------------------------------------------
__global__ __launch_bounds__(256) void per_batch_kernel(
    const float* __restrict__ H2, const float* __restrict__ W3,
    const float* __restrict__ b3, const float* __restrict__ P,
    float* __restrict__ knots_out, float* __restrict__ bd, int Bn) {
  const int b = blockIdx.x * blockDim.x + threadIdx.x;
  if (b >= Bn) return;

  // ---- layer 3: logits = H2[b] . W3^T + b3 ----
  const float* h = H2 + (size_t)b * 128;
  float l[5];
#pragma unroll
  for (int j = 0; j < 5; ++j) l[j] = b3[j];
  for (int k = 0; k < 128; ++k) {
    const float hv = h[k];
#pragma unroll
    for (int j = 0; j < 5; ++j) l[j] += hv * W3[j * 128 + k];
  }

  // ---- softmax over 5, cumsum of first 4 -> interior knots ----
  float mx = l[0];
#pragma unroll
  for (int j = 1; j < 5; ++j) mx = fmaxf(mx, l[j]);
  float e[5], s = 0.0f;
#pragma unroll
  for (int j = 0; j < 5; ++j) { e[j] = expf(l[j] - mx); s += e[j]; }
  const float inv = 1.0f / s;

  float kn[12];
#pragma unroll
  for (int j = 0; j < 4; ++j) kn[j] = 0.0f;
  float cum = 0.0f;
#pragma unroll
  for (int j = 0; j < 4; ++j) { cum += e[j] * inv; kn[4 + j] = cum; }
#pragma unroll
  for (int j = 8; j < 12; ++j) kn[j] = 1.0f;

#pragma unroll
  for (int j = 0; j < 12; ++j) knots_out[b * 12 + j] = kn[j];

  // ---- A matrix ----
  float A[8][8];
#pragma unroll
  for (int r = 0; r < 8; ++r)
#pragma unroll
    for (int j = 0; j < 8; ++j) A[r][j] = 0.0f;

  A[0][0] = kn[5];
  A[0][1] = -kn[4] - kn[5];
  A[0][2] = kn[4];
  float N[11];
  for (int r = 0; r < 5; ++r) {
    construct_row(kn[3 + r], kn, N);
#pragma unroll
    for (int j = 0; j < 8; ++j) A[1 + r][j] = N[j];
  }
  A[6][7] = 1.0f;
  A[7][5] = 1.0f - kn[7];
  A[7][6] = kn[7] + kn[6] - 2.0f;
  A[7][7] = 1.0f - kn[6];

  // ---- RHS: [0, P[b,:,d], 0] ----
  float bx[8], by[8];
  bx[0] = by[0] = 0.0f;
  bx[7] = by[7] = 0.0f;
#pragma unroll
  for (int i = 0; i < 6; ++i) {
    bx[1 + i] = P[b * 12 + i * 2 + 0];
    by[1 + i] = P[b * 12 + i * 2 + 1];
  }

  // ---- Gauss-Jordan with partial pivoting, 2 RHS ----
  for (int col = 0; col < 8; ++col) {
    int   piv  = col;
    float best = fabsf(A[col][col]);
    for (int r = col + 1; r < 8; ++r) {
      const float v = fabsf(A[r][col]);
      if (v > best) { best = v; piv = r; }
    }
    if (piv != col) {
#pragma unroll
      for (int j = 0; j < 8; ++j) {
        const float t = A[col][j]; A[col][j] = A[piv][j]; A[piv][j] = t;
      }
      float t = bx[col]; bx[col] = bx[piv]; bx[piv] = t;
      t = by[col]; by[col] = by[piv]; by[piv] = t;
    }
    const float dinv = 1.0f / A[col][col];
#pragma unroll
    for (int j = 0; j < 8; ++j) A[col][j] *= dinv;
    bx[col] *= dinv;
    by[col] *= dinv;
    for (int r = 0; r < 8; ++r) {
      if (r == col) continue;
      const float f = A[r][col];
#pragma unroll
      for (int j = 0; j < 8; ++j) A[r][j] -= f * A[col][j];
      bx[r] -= f * bx[col];
      by[r] -= f * by[col];
    }
  }

  // ---- first / second derivative control points ----
  float Qx[7], Qy[7], Rx[6], Ry[6];
#pragma unroll
  for (int i = 0; i < 7; ++i) {
    const float d = 3.0f / (kn[i + 4] - kn[i + 1]);
    Qx[i] = d * (bx[i + 1] - bx[i]);
    Qy[i] = d * (by[i + 1] - by[i]);
  }
#pragma unroll
  for (int i = 0; i < 6; ++i) {
    const float d = 2.0f / (kn[i + 4] - kn[i + 2]);
    Rx[i] = d * (Qx[i + 1] - Qx[i]);
    Ry[i] = d * (Qy[i + 1] - Qy[i]);
  }

  float* o = bd + (size_t)b * 40;
#pragma unroll
  for (int i = 0; i < 12; ++i) o[i] = kn[i];
#pragma unroll
  for (int i = 0; i < 7; ++i) { o[12 + i] = Qx[i]; o[19 + i] = Qy[i]; }
#pragma unroll
  for (int i = 0; i < 6; ++i) { o[26 + i] = Rx[i]; o[32 + i] = Ry[i]; }
}

// ---------------------------------------------------------------------------
// One wave per batch: evaluate de Boor (p=2 on Q, p=1 on R) at 1000 points,
// accumulate Simpson-weighted curvature energy, wave32 shuffle reduction.
// ---------------------------------------------------------------------------
__global__ __launch_bounds__(256) void integrate_kernel(
    const float* __restrict__ bd, float* __restrict__ out, int Bn) {
  const int lane = threadIdx.x & 31;
  const int wave = threadIdx.x >> 5;
  const int b    = blockIdx.x * (blockDim.x >> 5) + wave;
  if (b >= Bn) return;

  const float* d = bd + (size_t)b * 40;  // [kn(12)|Qx(7)|Qy(7)|Rx(6)|Ry(6)]

  float kn[12];
#pragma unroll
  for (int i = 0; i < 12; ++i) kn[i] = d[i];

  float acc = 0.0f;
  for (int i = lane; i < NUM_PTS; i += 32) {
    const float t = (float)i * (1.0f / (float)(NUM_PTS - 1));
    // k = #{interior knots < t} (same index for both derivative splines)
    int k = 0;
#pragma unroll
    for (int j = 0; j < 4; ++j) k += (t > kn[4 + j]) ? 1 : 0;

    const float a1 = kn[4 + k];  // dk[3+k]
    const float a2 = kn[2 + k];  // dk[1+k]
    const float a3 = kn[5 + k];  // dk[4+k]
    const float a4 = kn[3 + k];  // dk[2+k]

    const float qx0 = d[12 + k], qx1 = d[13 + k], qx2 = d[14 + k];
    const float qy0 = d[19 + k], qy1 = d[20 + k], qy2 = d[21 + k];
    const float rx0 = d[26 + k], rx1 = d[27 + k];
    const float ry0 = d[32 + k], ry1 = d[33 + k];

    // de Boor degree 2 (first derivative spline)
    const float i12 = 1.0f / (a1 - a2);
    const float i34 = 1.0f / (a3 - a4);
    const float i14 = 1.0f / (a1 - a4);
    const float g0x = ((a1 - t) * qx0 + (t - a2) * qx1) * i12;
    const float g1x = ((a3 - t) * qx1 + (t - a4) * qx2) * i34;
    const float g0y = ((a1 - t) * qy0 + (t - a2) * qy1) * i12;
    const float g1y = ((a3 - t) * qy1 + (t - a4) * qy2) * i34;
    const float spx = ((a1 - t) * g0x + (t - a4) * g1x) * i14;
    const float spy = ((a1 - t) * g0y + (t - a4) * g1y) * i14;

    // de Boor degree 1 (second derivative spline)
    const float sdx = ((a1 - t) * rx0 + (t - a4) * rx1) * i14;
    const float sdy = ((a1 - t) * ry0 + (t - a4) * ry1) * i14;

    const float speed2 = spx * spx + spy * spy;
    const float cross  = spx * sdy - spy * sdx;
    const float val    = (cross * cross) / (speed2 * speed2 * sqrtf(speed2));
    acc += ((i & 1) ? 4.0f : 2.0f) * val;
  }

#pragma unroll
  for (int off = 16; off > 0; off >>= 1) acc += __shfl_xor(acc, off, 32);
  if (lane == 0)
    out[b] = acc * ((1.0f / (float)(NUM_PTS - 1)) / 3.0f);
}

// ---------------------------------------------------------------------------
extern "C" void kernel_launch(void* const* d_in, const int* in_sizes, int n_in,
                              void* d_out, int out_size, void* d_ws,
                              size_t ws_size, hipStream_t stream) {
  const float* x  = (const float*)d_in[0];  // (B, 8)
  const float* P  = (const float*)d_in[1];  // (B, 6, 2, 1)
  const float* W1 = (const float*)d_in[2];  // (128, 8)
  const float* b1 = (const float*)d_in[3];  // (128,)
  const float* W2 = (const float*)d_in[4];  // (128, 128)
  const float* b2 = (const float*)d_in[5];  // (128,)
  const float* W3 = (const float*)d_in[6];  // (5, 128)
  const float* b3 = (const float*)d_in[7];  // (5,)

  const int B = in_sizes[0] / 8;

  float* ws = (float*)d_ws;
  float* H1 = ws;                        // B*128
  float* H2 = H1 + (size_t)B * 128;      // B*128
  float* bd = H2 + (size_t)B * 128;      // B*40

  float* out       = (float*)d_out;      // [integration(B) | knots(B*12)]
  float* knots_out = out + B;

  const int mtiles = (B + 15) / 16;
  mlp_layer_wmma<8><<<mtiles, 256, 0, stream>>>(x, W1, b1, H1, B);
  mlp_layer_wmma<128><<<mtiles, 256, 0, stream>>>(H1, W2, b2, H2, B);
  per_batch_kernel<<<(B + 255) / 256, 256, 0, stream>>>(H2, W3, b3, P,
                                                        knots_out, bd, B);
  const int wpb = 8;  // waves per block
  integrate_kernel<<<(B + wpb - 1) / wpb, wpb * 32, 0, stream>>>(bd, out, B);
}